// Wireless_Autoencoder_10196252360819
// MI455X (gfx1250) — compile-verified
//
#include <hip/hip_runtime.h>
#include <hip/hip_bf16.h>

// ---------------------------------------------------------------------------
// Wireless autoencoder, MI455X (gfx1250), wave32 + WMMA f32 16x16x4.
//
// Kernel 1 (enc): x -> elu(x@w1) -> elu(@w2) -> conv/tanh chain (scalar per
//   lane) -> tanh(@w3) -> @w4 = pre-batchnorm h. h staged to ws, per-block
//   partial sums (h, h^2) written to ws.
// Kernel 2 (stats): reduce partials -> mu[16], rsqrt(var+eps)[16].
// Kernel 3 (dec): normalize, 3-tap complex fading channel + noise, then
//   elu(@dw1) -> @dw2 via WMMA, write d_out.
//
// Each wave owns 32 samples (two 16-row WMMA M-tiles). Activations are the
// WMMA A operand, weights the B operand. LDS (per-wave 32x36 f32 tile,
// even row stride => 8B-aligned v2f A loads) provides the transposes
// between dependent GEMMs.
// ---------------------------------------------------------------------------

typedef float v2f __attribute__((ext_vector_type(2)));
typedef float v8f __attribute__((ext_vector_type(8)));

#define BATCH      524288
#define NBLK       2048          // BATCH / 256 samples per block
#define LSTRIDE    36            // LDS row stride in floats (even, padded)
#define NOISE_STD  0.4466835921509630f
#define INV_SQRT2  0.7071067811865476f

__device__ __forceinline__ float fast_tanh(float x) {
  float ax = __builtin_fabsf(x);
  float e  = __expf(-2.0f * ax);
  float r  = (1.0f - e) / (1.0f + e);
  return x < 0.0f ? -r : r;
}
__device__ __forceinline__ float elu1(float x) {
  return x > 0.0f ? x : __expf(x) - 1.0f;
}

__device__ __forceinline__ v8f wmma4(v2f a, v2f b, v8f c) {
  // D = A(16x4 f32) * B(4x16 f32) + C(16x16 f32)
  return __builtin_amdgcn_wmma_f32_16x16x4_f32(false, a, false, b,
                                               (short)0, c, false, false);
}

// ---------------------------------------------------------------------------
__global__ __launch_bounds__(256)
void enc_kernel(const float* __restrict__ x,
                const float* __restrict__ w1, const float* __restrict__ b1,
                const float* __restrict__ w2, const float* __restrict__ b2,
                const float* __restrict__ cw1, const float* __restrict__ cb1,
                const float* __restrict__ cw2, const float* __restrict__ cb2,
                const float* __restrict__ cw3, const float* __restrict__ cb3,
                const float* __restrict__ cw4, const float* __restrict__ cb4,
                const float* __restrict__ w3, const float* __restrict__ b3,
                const float* __restrict__ w4, const float* __restrict__ b4,
                float* __restrict__ hout, float* __restrict__ partials)
{
  __shared__ float lds[8][32 * LSTRIDE];
  __shared__ float bsum[16], bsq[16];

  const int tid  = threadIdx.x;
  const int lane = tid & 31;
  const int wave = tid >> 5;
  const int half = lane >> 4;      // 0: lanes 0-15, 1: lanes 16-31
  const int lm   = lane & 15;
  float* S = lds[wave];
  const int base = blockIdx.x * 256 + wave * 32;

  if (tid < 16) { bsum[tid] = 0.0f; bsq[tid] = 0.0f; }

  // ---------------- stage 1: h1 = elu(x @ w1 + b1) ----------------
  v2f bw1[4], bw2[4];
#pragma unroll
  for (int s = 0; s < 4; ++s) {
    int k = 4 * s + 2 * half;
    bw1[s].x = w1[(k    ) * 16 + lm];
    bw1[s].y = w1[(k + 1) * 16 + lm];
    bw2[s].x = w2[(k    ) * 16 + lm];
    bw2[s].y = w2[(k + 1) * 16 + lm];
  }
  const float bias1 = b1[lm], bias2 = b2[lm];

#pragma unroll
  for (int t = 0; t < 2; ++t) {
    v8f acc;
#pragma unroll
    for (int r = 0; r < 8; ++r) acc[r] = bias1;
#pragma unroll
    for (int s = 0; s < 4; ++s) {
      const float* ap = x + (size_t)(base + 16 * t + lm) * 16 + 4 * s + 2 * half;
      v2f a = *(const v2f*)ap;
      acc = wmma4(a, bw1[s], acc);
    }
#pragma unroll
    for (int r = 0; r < 8; ++r)
      S[(16 * t + r + 8 * half) * LSTRIDE + lm] = elu1(acc[r]);  // cols 0..15
  }
  __syncthreads();

  // ---------------- stage 2: h2 = elu(h1 @ w2 + b2) ----------------
#pragma unroll
  for (int t = 0; t < 2; ++t) {
    v8f acc;
#pragma unroll
    for (int r = 0; r < 8; ++r) acc[r] = bias2;
#pragma unroll
    for (int s = 0; s < 4; ++s) {
      v2f a = *(const v2f*)&S[(16 * t + lm) * LSTRIDE + 4 * s + 2 * half];
      acc = wmma4(a, bw2[s], acc);
    }
#pragma unroll
    for (int r = 0; r < 8; ++r)
      S[(16 * t + r + 8 * half) * LSTRIDE + 16 + lm] = elu1(acc[r]); // cols 16..31
  }
  __syncthreads();

  // ---------------- conv chain: one sample per lane ----------------
  {
    float v[16];
#pragma unroll
    for (int j = 0; j < 16; ++j) v[j] = S[lane * LSTRIDE + 16 + j];

    float y1[8][15];
#pragma unroll
    for (int c = 0; c < 8; ++c) {
      float k0 = cw1[2 * c], k1 = cw1[2 * c + 1], bb = cb1[c];
#pragma unroll
      for (int t = 0; t < 15; ++t)
        y1[c][t] = fast_tanh(k0 * v[t] + k1 * v[t + 1] + bb);
    }

    float y2[8][6];                       // kernel 4, stride 2
#pragma unroll
    for (int o = 0; o < 8; ++o) {
#pragma unroll
      for (int t = 0; t < 6; ++t) {
        float s = cb2[o];
#pragma unroll
        for (int i = 0; i < 8; ++i) {
          const float* wp = cw2 + (o * 8 + i) * 4;
#pragma unroll
          for (int k = 0; k < 4; ++k) s += wp[k] * y1[i][2 * t + k];
        }
        y2[o][t] = fast_tanh(s);
      }
    }

    float y3[8][5];                       // kernel 2, stride 1
#pragma unroll
    for (int o = 0; o < 8; ++o) {
#pragma unroll
      for (int t = 0; t < 5; ++t) {
        float s = cb3[o];
#pragma unroll
        for (int i = 0; i < 8; ++i) {
          const float* wp = cw3 + (o * 8 + i) * 2;
          s += wp[0] * y2[i][t] + wp[1] * y2[i][t + 1];
        }
        y3[o][t] = fast_tanh(s);
      }
    }

    float y4[8][4];                       // kernel 2, stride 1
#pragma unroll
    for (int o = 0; o < 8; ++o) {
#pragma unroll
      for (int t = 0; t < 4; ++t) {
        float s = cb4[o];
#pragma unroll
        for (int i = 0; i < 8; ++i) {
          const float* wp = cw4 + (o * 8 + i) * 2;
          s += wp[0] * y3[i][t] + wp[1] * y3[i][t + 1];
        }
        y4[o][t] = fast_tanh(s);
      }
    }

    // flat (channel-major) -> LDS cols 0..31 of this lane's row
#pragma unroll
    for (int o = 0; o < 8; ++o)
#pragma unroll
      for (int t = 0; t < 4; ++t)
        S[lane * LSTRIDE + o * 4 + t] = y4[o][t];
  }
  __syncthreads();

  // ---------------- stage 3: h3 = tanh(flat @ w3 + b3), N=32, K=32 --------
  v2f bw3[2][8], bw4[8];
#pragma unroll
  for (int s = 0; s < 8; ++s) {
    int k = 4 * s + 2 * half;
    bw3[0][s].x = w3[(k    ) * 32 + lm];
    bw3[0][s].y = w3[(k + 1) * 32 + lm];
    bw3[1][s].x = w3[(k    ) * 32 + 16 + lm];
    bw3[1][s].y = w3[(k + 1) * 32 + 16 + lm];
    bw4[s].x    = w4[(k    ) * 16 + lm];
    bw4[s].y    = w4[(k + 1) * 16 + lm];
  }
  const float bias3[2] = { b3[lm], b3[16 + lm] };
  const float bias4 = b4[lm];

  v8f h3[2][2];
#pragma unroll
  for (int t = 0; t < 2; ++t) {
#pragma unroll
    for (int j = 0; j < 2; ++j) {
      v8f acc;
#pragma unroll
      for (int r = 0; r < 8; ++r) acc[r] = bias3[j];
#pragma unroll
      for (int s = 0; s < 8; ++s) {
        v2f a = *(const v2f*)&S[(16 * t + lm) * LSTRIDE + 4 * s + 2 * half];
        acc = wmma4(a, bw3[j][s], acc);
      }
#pragma unroll
      for (int r = 0; r < 8; ++r) acc[r] = fast_tanh(acc[r]);
      h3[t][j] = acc;
    }
  }
  __syncthreads();
#pragma unroll
  for (int t = 0; t < 2; ++t)
#pragma unroll
    for (int j = 0; j < 2; ++j)
#pragma unroll
      for (int r = 0; r < 8; ++r)
        S[(16 * t + r + 8 * half) * LSTRIDE + 16 * j + lm] = h3[t][j][r];
  __syncthreads();

  // ---------------- stage 4: h = h3 @ w4 + b4 (pre-batchnorm) -------------
  float psum = 0.0f, psq = 0.0f;
#pragma unroll
  for (int t = 0; t < 2; ++t) {
    v8f acc;
#pragma unroll
    for (int r = 0; r < 8; ++r) acc[r] = bias4;
#pragma unroll
    for (int s = 0; s < 8; ++s) {
      v2f a = *(const v2f*)&S[(16 * t + lm) * LSTRIDE + 4 * s + 2 * half];
      acc = wmma4(a, bw4[s], acc);
    }
#pragma unroll
    for (int r = 0; r < 8; ++r) {
      float val = acc[r];
      hout[(size_t)(base + 16 * t + r + 8 * half) * 16 + lm] = val;
      psum += val;
      psq  += val * val;
    }
  }
  atomicAdd(&bsum[lm], psum);
  atomicAdd(&bsq[lm], psq);
  __syncthreads();
  if (tid < 32)
    partials[(size_t)blockIdx.x * 32 + tid] = (tid < 16) ? bsum[tid] : bsq[tid - 16];
}

// ---------------------------------------------------------------------------
__global__ __launch_bounds__(256)
void stats_kernel(const float* __restrict__ partials, float* __restrict__ stats)
{
  __shared__ float red[8][32];
  __shared__ float tot[32];
  const int tid = threadIdx.x;
  const int col = tid & 31, seg = tid >> 5;
  float s = 0.0f;
  for (int r = seg * 256; r < seg * 256 + 256; ++r)
    s += partials[(size_t)r * 32 + col];
  red[seg][col] = s;
  __syncthreads();
  if (tid < 32) {
    float t = 0.0f;
#pragma unroll
    for (int k = 0; k < 8; ++k) t += red[k][tid];
    tot[tid] = t;
  }
  __syncthreads();
  if (tid < 16) {
    const float invB = 1.0f / (float)BATCH;
    float mu  = tot[tid] * invB;
    float msq = tot[tid + 16] * invB;
    float var = msq - mu * mu;
    stats[tid]      = mu;
    stats[16 + tid] = rsqrtf(var + 1e-5f);
  }
}

// ---------------------------------------------------------------------------
__global__ __launch_bounds__(256)
void dec_kernel(const float* __restrict__ hin, const float* __restrict__ stats,
                const float* __restrict__ noise, const float* __restrict__ fading,
                const float* __restrict__ dw1, const float* __restrict__ db1,
                const float* __restrict__ dw2, const float* __restrict__ db2,
                float* __restrict__ out)
{
  __shared__ float lds[8][32 * LSTRIDE];
  __shared__ float sst[32];

  const int tid  = threadIdx.x;
  const int lane = tid & 31;
  const int wave = tid >> 5;
  const int half = lane >> 4;
  const int lm   = lane & 15;
  float* S = lds[wave];
  const int base = blockIdx.x * 256 + wave * 32;
  const int sidx = base + lane;

  if (tid < 32) sst[tid] = stats[tid];
  __syncthreads();

  // normalize
  float e[16];
#pragma unroll
  for (int j = 0; j < 16; ++j) {
    float h = hin[(size_t)sidx * 16 + j];
    e[j] = (h - sst[j]) * sst[16 + j];
  }

  // 3-tap complex fading channel + AWGN
  float xre[8], xim[8];
#pragma unroll
  for (int c = 0; c < 8; ++c) { xre[c] = e[2 * c]; xim[c] = e[2 * c + 1]; }
  float hre[3], him[3];
#pragma unroll
  for (int t = 0; t < 3; ++t) {
    hre[t] = fading[(size_t)sidx * 6 + 2 * t    ] * INV_SQRT2;
    him[t] = fading[(size_t)sidx * 6 + 2 * t + 1] * INV_SQRT2;
  }
  float cvec[16];
#pragma unroll
  for (int n = 0; n < 8; ++n) {
    float ore = 0.0f, oim = 0.0f;
#pragma unroll
    for (int t = 0; t < 3; ++t) {
      int idx = n - t;
      if (idx >= 0) {
        ore += hre[t] * xre[idx] - him[t] * xim[idx];
        oim += hre[t] * xim[idx] + him[t] * xre[idx];
      }
    }
    cvec[2 * n    ] = ore + noise[(size_t)sidx * 16 + 2 * n    ] * NOISE_STD;
    cvec[2 * n + 1] = oim + noise[(size_t)sidx * 16 + 2 * n + 1] * NOISE_STD;
  }
#pragma unroll
  for (int j = 0; j < 16; ++j) S[lane * LSTRIDE + j] = cvec[j];
  __syncthreads();

  // decoder weights as WMMA B operands
  v2f bd1[2][4], bd2[8];
#pragma unroll
  for (int s = 0; s < 4; ++s) {
    int k = 4 * s + 2 * half;
    bd1[0][s].x = dw1[(k    ) * 32 + lm];
    bd1[0][s].y = dw1[(k + 1) * 32 + lm];
    bd1[1][s].x = dw1[(k    ) * 32 + 16 + lm];
    bd1[1][s].y = dw1[(k + 1) * 32 + 16 + lm];
  }
#pragma unroll
  for (int s = 0; s < 8; ++s) {
    int k = 4 * s + 2 * half;
    bd2[s].x = dw2[(k    ) * 16 + lm];
    bd2[s].y = dw2[(k + 1) * 16 + lm];
  }
  const float biasd1[2] = { db1[lm], db1[16 + lm] };
  const float biasd2 = db2[lm];

  // d = elu(c @ dw1 + db1), N=32, K=16
  v8f dacc[2][2];
#pragma unroll
  for (int t = 0; t < 2; ++t) {
#pragma unroll
    for (int j = 0; j < 2; ++j) {
      v8f acc;
#pragma unroll
      for (int r = 0; r < 8; ++r) acc[r] = biasd1[j];
#pragma unroll
      for (int s = 0; s < 4; ++s) {
        v2f a = *(const v2f*)&S[(16 * t + lm) * LSTRIDE + 4 * s + 2 * half];
        acc = wmma4(a, bd1[j][s], acc);
      }
#pragma unroll
      for (int r = 0; r < 8; ++r) acc[r] = elu1(acc[r]);
      dacc[t][j] = acc;
    }
  }
  __syncthreads();
#pragma unroll
  for (int t = 0; t < 2; ++t)
#pragma unroll
    for (int j = 0; j < 2; ++j)
#pragma unroll
      for (int r = 0; r < 8; ++r)
        S[(16 * t + r + 8 * half) * LSTRIDE + 16 * j + lm] = dacc[t][j][r];
  __syncthreads();

  // out = d @ dw2 + db2, N=16, K=32
#pragma unroll
  for (int t = 0; t < 2; ++t) {
    v8f acc;
#pragma unroll
    for (int r = 0; r < 8; ++r) acc[r] = biasd2;
#pragma unroll
    for (int s = 0; s < 8; ++s) {
      v2f a = *(const v2f*)&S[(16 * t + lm) * LSTRIDE + 4 * s + 2 * half];
      acc = wmma4(a, bd2[s], acc);
    }
#pragma unroll
    for (int r = 0; r < 8; ++r)
      out[(size_t)(base + 16 * t + r + 8 * half) * 16 + lm] = acc[r];
  }
}

// ---------------------------------------------------------------------------
extern "C" void kernel_launch(void* const* d_in, const int* in_sizes, int n_in,
                              void* d_out, int out_size, void* d_ws, size_t ws_size,
                              hipStream_t stream)
{
  const float* x      = (const float*)d_in[0];
  const float* noise  = (const float*)d_in[1];
  const float* fading = (const float*)d_in[2];
  const float* w1  = (const float*)d_in[3];
  const float* b1  = (const float*)d_in[4];
  const float* w2  = (const float*)d_in[5];
  const float* b2  = (const float*)d_in[6];
  const float* cw1 = (const float*)d_in[7];
  const float* cb1 = (const float*)d_in[8];
  const float* cw2 = (const float*)d_in[9];
  const float* cb2 = (const float*)d_in[10];
  const float* cw3 = (const float*)d_in[11];
  const float* cb3 = (const float*)d_in[12];
  const float* cw4 = (const float*)d_in[13];
  const float* cb4 = (const float*)d_in[14];
  const float* w3  = (const float*)d_in[15];
  const float* b3  = (const float*)d_in[16];
  const float* w4  = (const float*)d_in[17];
  const float* b4  = (const float*)d_in[18];
  const float* dw1 = (const float*)d_in[19];
  const float* db1 = (const float*)d_in[20];
  const float* dw2 = (const float*)d_in[21];
  const float* db2 = (const float*)d_in[22];

  float* ws       = (float*)d_ws;
  float* hout     = ws;                              // BATCH*16 floats
  float* partials = ws + (size_t)BATCH * 16;         // NBLK*32 floats
  float* stats    = partials + (size_t)NBLK * 32;    // 32 floats

  enc_kernel<<<NBLK, 256, 0, stream>>>(x, w1, b1, w2, b2,
                                       cw1, cb1, cw2, cb2, cw3, cb3, cw4, cb4,
                                       w3, b3, w4, b4, hout, partials);
  stats_kernel<<<1, 256, 0, stream>>>(partials, stats);
  dec_kernel<<<NBLK, 256, 0, stream>>>(hout, stats, noise, fading,
                                       dw1, db1, dw2, db2, (float*)d_out);
}